// VINNOpenLoop_28114855919804
// MI455X (gfx1250) — compile-verified
//
#include <hip/hip_runtime.h>
#include <hip/hip_bf16.h>
#include <math.h>

// ---------------------------------------------------------------------------
// kNN (cdist -> top-16 -> softmax -> action blend), B=64, D=512, A=7, k=16
// GEMM via V_WMMA_F32_16X16X4_F32; staging via GLOBAL_LOAD_ASYNC_TO_LDS_B128
// with double-buffered prefetch (ASYNCcnt pipeline) on gfx1250.
// ---------------------------------------------------------------------------

#define NBLK    256     // blocks in kernel 1
#define TPB     256     // threads per block (8 wave32)
#define TILE_N  256     // columns per tile
#define KC      32      // K-chunk staged in LDS
#define BROWS   64      // batch rows (B)
#define DDIM    512     // feature dim
#define ADIM    7       // action dim
#define KNEI    16      // k
#define RSTR    36      // row stride (floats): 144B, 16B-aligned, conflict-free b64
#define DSTR    257     // d2 tile stride (floats), padded
#define BIGF    3.0e38f

// shared-memory partition (float offsets)
#define ASZ     (BROWS * RSTR)            // 2304
#define BSZ     (TILE_N * RSTR)           // 9216
#define A_OFF(b) ((b) * ASZ)              // 0 / 2304
#define B_OFF(b) (2 * ASZ + (b) * BSZ)    // 4608 / 13824
#define D2_OFF   (2 * ASZ + 2 * BSZ)      // 23040
#define XN_OFF   (D2_OFF + BROWS * DSTR)  // 39488
#define YN_OFF   (XN_OFF + BROWS)         // 39552
#define S_TOTAL  (YN_OFF + TILE_N)        // 39808 floats = 159232 B

typedef float v2f __attribute__((ext_vector_type(2)));
typedef float v8f __attribute__((ext_vector_type(8)));

// Async global->LDS 16B copy (GVS addressing: SGPR64 base + VGPR i32 offset).
// Tracked by ASYNCcnt; no VGPR data path.
__device__ __forceinline__ void async_ld16(unsigned lds_byte, int gbyte_off,
                                           const float* sbase) {
    asm volatile("global_load_async_to_lds_b128 %0, %1, %2"
                 :: "v"(lds_byte), "v"(gbyte_off), "s"(sbase)
                 : "memory");
}

__device__ __forceinline__ void wait_async0() {
    asm volatile("s_wait_asynccnt 0x0" ::: "memory");
}

__launch_bounds__(TPB)
__global__ void knn_gemm_topk(const float* __restrict__ batch,
                              const float* __restrict__ reps,
                              float* __restrict__ cand_val,
                              int*   __restrict__ cand_idx,
                              int N, int tilesPerBlk)
{
    __shared__ float S[S_TOTAL];
    float* D2 = &S[D2_OFF];
    float* Xn = &S[XN_OFF];
    float* Yn = &S[YN_OFF];

    const int tid  = threadIdx.x;
    const int wave = tid >> 5;
    const int lane = tid & 31;
    const int half = lane >> 4;
    const int lq   = lane & 15;

    // per-thread running top-16 (sorted ascending), in registers
    float tv[KNEI];
    int   ti[KNEI];
#pragma unroll
    for (int q = 0; q < KNEI; ++q) { tv[q] = BIGF; ti[q] = -1; }

    // staging geometry: each thread owns fixed (row,seg) pairs; 16B per async op
    // A: 64 rows x 8 segs  = 512 ops  -> 2/thread
    // B: 256 rows x 8 segs = 2048 ops -> 8/thread

    for (int tile = 0; tile < tilesPerBlk; ++tile) {
        const int col0 = (blockIdx.x * tilesPerBlk + tile) * TILE_N;

        v8f c[4][2];
#pragma unroll
        for (int mt = 0; mt < 4; ++mt)
#pragma unroll
            for (int nt = 0; nt < 2; ++nt)
                c[mt][nt] = (v8f)0.0f;

        float xacc = 0.0f;
        float yacc = 0.0f;

        // ---- prologue: stage chunk 0 into buffer 0 ----
        {
#pragma unroll
            for (int i = 0; i < 2; ++i) {
                int f = i * TPB + tid, row = f >> 3, seg = f & 7;
                unsigned lds = (unsigned)(uintptr_t)&S[A_OFF(0) + row * RSTR + seg * 4];
                async_ld16(lds, (row * DDIM + seg * 4) * 4, batch);
            }
#pragma unroll
            for (int i = 0; i < 8; ++i) {
                int f = i * TPB + tid, row = f >> 3, seg = f & 7;
                int gcol = col0 + row; if (gcol >= N) gcol = N - 1;
                unsigned lds = (unsigned)(uintptr_t)&S[B_OFF(0) + row * RSTR + seg * 4];
                async_ld16(lds, (gcol * DDIM + seg * 4) * 4, reps);
            }
        }

        int buf = 0;
        for (int kc = 0; kc < DDIM; kc += KC) {
            wait_async0();        // current buffer's copies landed (this wave)
            __syncthreads();      // ...for all waves; prev buffer fully consumed

            // ---- prefetch next chunk into the other buffer ----
            if (kc + KC < DDIM) {
                const int kn = kc + KC;
                const int nb = buf ^ 1;
#pragma unroll
                for (int i = 0; i < 2; ++i) {
                    int f = i * TPB + tid, row = f >> 3, seg = f & 7;
                    unsigned lds = (unsigned)(uintptr_t)&S[A_OFF(nb) + row * RSTR + seg * 4];
                    async_ld16(lds, (row * DDIM + kn + seg * 4) * 4, batch);
                }
#pragma unroll
                for (int i = 0; i < 8; ++i) {
                    int f = i * TPB + tid, row = f >> 3, seg = f & 7;
                    int gcol = col0 + row; if (gcol >= N) gcol = N - 1;
                    unsigned lds = (unsigned)(uintptr_t)&S[B_OFF(nb) + row * RSTR + seg * 4];
                    async_ld16(lds, (gcol * DDIM + kn + seg * 4) * 4, reps);
                }
            }

            const float* Ab = &S[A_OFF(buf)];
            const float* Bb = &S[B_OFF(buf)];

            // ---- norm partials (row-major reads) ----
            if (tid < BROWS) {
#pragma unroll
                for (int kk = 0; kk < KC; ++kk) {
                    float a = Ab[tid * RSTR + kk];
                    xacc += a * a;
                }
            }
#pragma unroll
            for (int kk = 0; kk < KC; ++kk) {
                float b = Bb[tid * RSTR + kk];
                yacc += b * b;
            }

            // ---- WMMA: 8 k-steps x (4 M-tiles x 2 N-tiles) ----
#pragma unroll
            for (int kk = 0; kk < KC; kk += 4) {
                v2f bfrag[2];
#pragma unroll
                for (int nt = 0; nt < 2; ++nt) {
                    int n = wave * 32 + nt * 16 + lq;
                    bfrag[nt] = *(const v2f*)&Bb[n * RSTR + kk + 2 * half];
                }
#pragma unroll
                for (int mt = 0; mt < 4; ++mt) {
                    int m = mt * 16 + lq;
                    v2f afrag = *(const v2f*)&Ab[m * RSTR + kk + 2 * half];
#pragma unroll
                    for (int nt = 0; nt < 2; ++nt) {
                        c[mt][nt] = __builtin_amdgcn_wmma_f32_16x16x4_f32(
                            false, afrag, false, bfrag[nt],
                            (short)0, c[mt][nt], false, false);
                    }
                }
            }
            buf ^= 1;
        } // kc

        if (tid < BROWS) Xn[tid] = xacc;
        Yn[tid] = yacc;
        __syncthreads();

        // ---- combine: d2 = ||x||^2 - 2 x.y + ||y||^2 -> padded LDS tile ----
#pragma unroll
        for (int mt = 0; mt < 4; ++mt) {
#pragma unroll
            for (int nt = 0; nt < 2; ++nt) {
                int n = wave * 32 + nt * 16 + lq;
                float yn = Yn[n];
#pragma unroll
                for (int r = 0; r < 8; ++r) {
                    int m = mt * 16 + half * 8 + r;   // C layout: VGPR r -> M=r / r+8
                    D2[m * DSTR + n] = Xn[m] - 2.0f * c[mt][nt][r] + yn;
                }
            }
        }
        __syncthreads();

        // ---- selection: 4 threads/row, each scans 64 cols, register top-16 ----
        {
            int m = tid & 63;
            int g = tid >> 6;
            for (int j = 0; j < 64; ++j) {
                int n    = g * 64 + j;
                int gcol = col0 + n;
                if (gcol >= N) break;          // cols ascending -> safe
                float v = D2[m * DSTR + n];
                if (v < tv[KNEI - 1]) {
                    tv[KNEI - 1] = v;
                    ti[KNEI - 1] = gcol;
#pragma unroll
                    for (int q = KNEI - 1; q > 0; --q) {
                        if (tv[q] < tv[q - 1]) {
                            float fv = tv[q]; tv[q] = tv[q - 1]; tv[q - 1] = fv;
                            int   iv = ti[q]; ti[q] = ti[q - 1]; ti[q - 1] = iv;
                        } else break;
                    }
                }
            }
        }
        __syncthreads();   // selection done before next tile restages LDS
    } // tile

    // ---- per-block merge: 4x16 candidates/row -> block top-16 (overlay on D2)
    {
        float* ov = D2;                                    // 64 x 65 vals
        int*   oi = (int*)&D2[64 * 65 + 32];               // 64 x 65 idx
        int m = tid & 63;
        int g = tid >> 6;
#pragma unroll
        for (int q = 0; q < KNEI; ++q) {
            ov[m * 65 + g * 16 + q] = tv[q];
            oi[m * 65 + g * 16 + q] = ti[q];
        }
        __syncthreads();
        if (tid < BROWS) {
            int m2 = tid;
            for (int p = 0; p < KNEI; ++p) {
                float best = BIGF; int bj = -1;
                for (int j = 0; j < 64; ++j) {
                    float v = ov[m2 * 65 + j];
                    if (v < best) { best = v; bj = j; }
                }
                int bidx = (bj >= 0) ? oi[m2 * 65 + bj] : -1;
                if (bj >= 0) ov[m2 * 65 + bj] = BIGF;
                cand_val[(blockIdx.x * BROWS + m2) * KNEI + p] = best;
                cand_idx[(blockIdx.x * BROWS + m2) * KNEI + p] = bidx;
            }
        }
    }
}

__launch_bounds__(TPB)
__global__ void knn_finalize(const float* __restrict__ cand_val,
                             const int*   __restrict__ cand_idx,
                             const float* __restrict__ actions,
                             float* __restrict__ out)
{
    __shared__ float lv[NBLK * KNEI];   // 4096
    __shared__ int   li[NBLK * KNEI];
    __shared__ float rv[TPB];
    __shared__ int   rp[TPB];
    __shared__ float topv[KNEI];
    __shared__ int   topi[KNEI];

    const int m   = blockIdx.x;
    const int tid = threadIdx.x;
    const int total = NBLK * KNEI;

    for (int i = tid; i < total; i += TPB) {
        int b = i >> 4, p = i & 15;
        lv[i] = cand_val[(b * BROWS + m) * KNEI + p];
        li[i] = cand_idx[(b * BROWS + m) * KNEI + p];
    }
    __syncthreads();

    for (int p = 0; p < KNEI; ++p) {
        float best = BIGF; int bpos = -1;
        for (int i = tid; i < total; i += TPB) {
            float v = lv[i];
            if (v < best) { best = v; bpos = i; }
        }
        rv[tid] = best; rp[tid] = bpos;
        __syncthreads();
        for (int s = TPB / 2; s > 0; s >>= 1) {
            if (tid < s) {
                if (rv[tid + s] < rv[tid]) { rv[tid] = rv[tid + s]; rp[tid] = rp[tid + s]; }
            }
            __syncthreads();
        }
        if (tid == 0) {
            int pos = rp[0];
            topv[p] = rv[0];
            topi[p] = (pos >= 0) ? li[pos] : -1;
            if (pos >= 0) lv[pos] = BIGF;
        }
        __syncthreads();
    }

    if (tid == 0) {
        float dist[KNEI], w[KNEI];
        float mn = BIGF;
#pragma unroll
        for (int q = 0; q < KNEI; ++q) {
            dist[q] = sqrtf(fmaxf(topv[q], 0.0f));
            mn = fminf(mn, dist[q]);
        }
        float ssum = 0.0f;
#pragma unroll
        for (int q = 0; q < KNEI; ++q) {
            w[q] = __expf(mn - dist[q]);   // softmax(-dist), stable
            ssum += w[q];
        }
        float inv = 1.0f / ssum;
#pragma unroll
        for (int a = 0; a < ADIM; ++a) {
            float acc = 0.0f;
#pragma unroll
            for (int q = 0; q < KNEI; ++q) {
                int idx = topi[q];
                acc += w[q] * actions[idx * ADIM + a];
            }
            out[m * ADIM + a] = acc * inv;
        }
        // indices follow pred in the flat output buffer
#pragma unroll
        for (int q = 0; q < KNEI; ++q)
            out[BROWS * ADIM + m * KNEI + q] = (float)topi[q];
    }
}

extern "C" void kernel_launch(void* const* d_in, const int* in_sizes, int n_in,
                              void* d_out, int out_size, void* d_ws, size_t ws_size,
                              hipStream_t stream) {
    const float* batch   = (const float*)d_in[0];   // [64, 512]
    const float* reps    = (const float*)d_in[1];   // [N, 512]
    const float* actions = (const float*)d_in[2];   // [N, 7]
    const int N = in_sizes[1] / DDIM;
    const int tilesPerBlk = (N + NBLK * TILE_N - 1) / (NBLK * TILE_N);

    float* cand_val = (float*)d_ws;                               // NBLK*64*16 f
    int*   cand_idx = (int*)d_ws + (size_t)NBLK * BROWS * KNEI;   // NBLK*64*16 i

    knn_gemm_topk<<<NBLK, TPB, 0, stream>>>(batch, reps, cand_val, cand_idx,
                                            N, tilesPerBlk);
    knn_finalize<<<BROWS, TPB, 0, stream>>>(cand_val, cand_idx, actions,
                                            (float*)d_out);
}